// Decomp_40707700032406
// MI455X (gfx1250) — compile-verified
//
#include <hip/hip_runtime.h>

// DEMA decomposition: res = x - ma, ma = Holt double-exponential smoothing.
// x: (32, 4096, 128) f32. Serial scan re-expressed as blocked affine scan:
//   per-16-step chunk: s_out = T(16x16 Toeplitz) @ X(16x16) + P @ z_start
// which maps onto v_wmma_f32_16x16x4_f32 (5 accumulating WMMAs per tile).
// Pass 3 stages the X tile through LDS so x is read from HBM exactly once there.

static constexpr double ALPHA = 0.3;
static constexpr double BETA  = 0.1;

static constexpr int Bn = 32, Tn = 4096, Cn = 128;
static constexpr int CHUNK  = 16;
static constexpr int NCHUNK = Tn / CHUNK;             // 256 (chunk 0 handled serially)
static constexpr size_t NELEM = (size_t)Bn * Tn * Cn; // 16,777,216

typedef float v2f __attribute__((ext_vector_type(2)));
typedef float v8f __attribute__((ext_vector_type(8)));

struct DemaConsts {
  float T[256];   // T[i*16+j] = (j<=i) ? cs[i-j] : 0   (A-matrix for WMMA)
  float cs[16];   // (M^k v)[0]
  float cp[16];   // (M^k v)[1]
  float Ps[16];   // (M^{i+1})[0][0]  carry coefficients
  float Pb[16];   // (M^{i+1})[0][1]
  float m16[4];   // M^16 row-major (chunk carry composition)
};

// ---------- Pass 1: per-chunk input-driven state increment u = sum M^{15-j} v x_j
__global__ __launch_bounds__(128) void dema_pass1(const float* __restrict__ x,
                                                  float* __restrict__ u,
                                                  DemaConsts cn) {
  const int k = blockIdx.x + 1;          // chunk 1..255
  const int b = blockIdx.y;
  const int c = threadIdx.x;             // channel
  const float* xp = x + ((size_t)b * Tn + (size_t)k * CHUNK) * Cn + c;
  float u0 = 0.f, u1 = 0.f;
#pragma unroll
  for (int j = 0; j < 16; ++j) {
    float xv = xp[(size_t)j * Cn];       // 512B coalesced per block per j
    u0 = fmaf(cn.cs[15 - j], xv, u0);
    u1 = fmaf(cn.cp[15 - j], xv, u1);
  }
  size_t idx = (((size_t)b * NCHUNK + k) * Cn + c) * 2;
  u[idx] = u0;
  u[idx + 1] = u1;
}

// ---------- Pass 2: serial head (t=0..15) + chunk-carry scan, store chunk start states
__global__ __launch_bounds__(128) void dema_pass2(const float* __restrict__ x,
                                                  float* __restrict__ res,
                                                  float* __restrict__ ma,
                                                  const float* __restrict__ u,
                                                  float* __restrict__ zst,
                                                  DemaConsts cn) {
  const int b = blockIdx.x;
  const int c = threadIdx.x;
  const float* xp = x   + (size_t)b * Tn * Cn + c;
  float* rp       = res + (size_t)b * Tn * Cn + c;
  float* mp       = ma  + (size_t)b * Tn * Cn + c;

  const float alpha = (float)ALPHA, beta = (float)BETA;
  float s  = xp[0];
  float bb = xp[Cn] - s;                 // b0 = x1 - s0
  mp[0] = s;
  rp[0] = 0.f;                           // x0 - s0 == 0
#pragma unroll
  for (int t = 1; t < 16; ++t) {
    float xv = xp[(size_t)t * Cn];
    float sn = fmaf(alpha, xv, (1.f - alpha) * (s + bb));
    bb = fmaf(beta, sn - s, (1.f - beta) * bb);
    s = sn;
    mp[(size_t)t * Cn] = s;
    rp[(size_t)t * Cn] = xv - s;
  }
  // s,bb == z_15 == start state of chunk 1
  const float m00 = cn.m16[0], m01 = cn.m16[1], m10 = cn.m16[2], m11 = cn.m16[3];
  for (int k = 1; k < NCHUNK; ++k) {
    size_t idx = (((size_t)b * NCHUNK + k) * Cn + c) * 2;
    zst[idx]     = s;
    zst[idx + 1] = bb;
    float u0 = u[idx], u1 = u[idx + 1];
    float sn = fmaf(m00, s, fmaf(m01, bb, u0));
    float bn = fmaf(m10, s, fmaf(m11, bb, u1));
    s = sn;
    bb = bn;
  }
}

// ---------- Pass 3: one wave per 16(time)x16(channel) tile, 5x v_wmma_f32_16x16x4_f32
// X tile loaded from HBM once (D-layout, kept in regs for the residual), staged
// through LDS to produce the B fragments (conflict-free: 32 distinct banks).
__global__ __launch_bounds__(256) void dema_pass3(const float* __restrict__ x,
                                                  float* __restrict__ res,
                                                  float* __restrict__ ma,
                                                  const float* __restrict__ zst,
                                                  DemaConsts cn) {
  __shared__ float xt[8 * 256];          // 16x16 tile per wave, 8 waves/block
  const int wave = (int)((blockIdx.x * blockDim.x + threadIdx.x) >> 5);
  const int wib  = threadIdx.x >> 5;     // wave-in-block
  const int lane = threadIdx.x & 31;
  const int l16  = lane & 15;
  const int hi   = lane >> 4;            // 0: lanes 0-15, 1: lanes 16-31
  const int ct = wave & 7;               // channel tile 0..7
  const int k  = ((wave >> 3) % (NCHUNK - 1)) + 1;   // chunk 1..255
  const int b  = wave / (8 * (NCHUNK - 1));
  const int c0 = ct * 16;

  const float* xbase = x + ((size_t)b * Tn + (size_t)k * CHUNK) * Cn + c0;
  float* tile = &xt[wib * 256];

  // Load tile in D-layout (row = r + 8*hi, col = l16): 64B segments per half-wave.
  float xv[8];
#pragma unroll
  for (int r = 0; r < 8; ++r) {
    const int mrow = r + hi * 8;
    xv[r] = xbase[(size_t)mrow * Cn + l16];
    tile[mrow * 16 + l16] = xv[r];       // same-wave LDS: DS ops are in-order
  }

  v8f acc = {};
  // D[i][n] += sum_j T[i][j] * X[j][n], K split into 4 slices of 4
#pragma unroll
  for (int q = 0; q < 4; ++q) {
    const int ka = 4 * q + 2 * hi;       // A: vgpr0=K ka, vgpr1=K ka+1 (per ISA layout)
    v2f A;
    A.x = cn.T[l16 * 16 + ka];
    A.y = cn.T[l16 * 16 + ka + 1];
    v2f Bm;                              // B rows ka, ka+1 striped on lanes, from LDS
    Bm.x = tile[ka * 16 + l16];
    Bm.y = tile[(ka + 1) * 16 + l16];
    acc = __builtin_amdgcn_wmma_f32_16x16x4_f32(false, A, false, Bm,
                                                (short)0, acc, false, false);
  }
  // carry: D[i][n] += Ps[i]*zs[n] + Pb[i]*zb[n]  (rank-2 update as a 5th WMMA)
  // Unconditional loads + selects: no EXEC divergence around the WMMA sequence.
  {
    size_t zi = (((size_t)b * NCHUNK + k) * Cn + c0 + l16) * 2;
    float zs = zst[zi];
    float zb = zst[zi + 1];
    float ps = cn.Ps[l16];
    float pb = cn.Pb[l16];
    v2f A, Bm;
    A.x  = hi ? 0.f : ps;
    A.y  = hi ? 0.f : pb;
    Bm.x = hi ? 0.f : zs;
    Bm.y = hi ? 0.f : zb;
    acc = __builtin_amdgcn_wmma_f32_16x16x4_f32(false, A, false, Bm,
                                                (short)0, acc, false, false);
  }
  // D layout: VGPR r -> row r (lanes 0-15) / row r+8 (lanes 16-31), col = lane%16.
  // Residual uses the register-resident x values: no second HBM read of x.
#pragma unroll
  for (int r = 0; r < 8; ++r) {
    const int mrow = r + hi * 8;
    size_t oi = ((size_t)b * Tn + (size_t)k * CHUNK + mrow) * Cn + c0 + l16;
    float mv = acc[r];
    ma[oi]  = mv;
    res[oi] = xv[r] - mv;
  }
}

extern "C" void kernel_launch(void* const* d_in, const int* in_sizes, int n_in,
                              void* d_out, int out_size, void* d_ws, size_t ws_size,
                              hipStream_t stream) {
  (void)in_sizes; (void)n_in; (void)out_size; (void)ws_size;
  const float* x = (const float*)d_in[0];
  float* res = (float*)d_out;            // output 0
  float* ma  = res + NELEM;              // output 1 (tuple order: res, ma)
  float* u   = (float*)d_ws;                                   // 8 MiB
  float* zst = u + (size_t)Bn * NCHUNK * Cn * 2;               // 8 MiB

  // Host-side constant generation (double precision), passed by value in kernarg.
  DemaConsts cn;
  const double a = ALPHA, be = BETA;
  const double M[2][2] = {{1.0 - a, 1.0 - a}, {-a * be, 1.0 - a * be}};
  double w0 = a, w1 = a * be;            // v, then M^k v
  for (int kk = 0; kk < 16; ++kk) {
    cn.cs[kk] = (float)w0;
    cn.cp[kk] = (float)w1;
    double n0 = M[0][0] * w0 + M[0][1] * w1;
    double n1 = M[1][0] * w0 + M[1][1] * w1;
    w0 = n0; w1 = n1;
  }
  double P[2][2] = {{M[0][0], M[0][1]}, {M[1][0], M[1][1]}};   // M^1
  for (int i = 0; i < 16; ++i) {
    cn.Ps[i] = (float)P[0][0];           // (M^{i+1})[0][0]
    cn.Pb[i] = (float)P[0][1];
    if (i < 15) {
      double Q[2][2];
      for (int r = 0; r < 2; ++r)
        for (int c = 0; c < 2; ++c)
          Q[r][c] = M[r][0] * P[0][c] + M[r][1] * P[1][c];
      P[0][0] = Q[0][0]; P[0][1] = Q[0][1]; P[1][0] = Q[1][0]; P[1][1] = Q[1][1];
    }
  }
  // P == M^16 now
  cn.m16[0] = (float)P[0][0]; cn.m16[1] = (float)P[0][1];
  cn.m16[2] = (float)P[1][0]; cn.m16[3] = (float)P[1][1];
  for (int i = 0; i < 16; ++i)
    for (int j = 0; j < 16; ++j)
      cn.T[i * 16 + j] = (j <= i) ? cn.cs[i - j] : 0.f;

  dema_pass1<<<dim3(NCHUNK - 1, Bn), 128, 0, stream>>>(x, u, cn);
  dema_pass2<<<Bn, 128, 0, stream>>>(x, res, ma, u, zst, cn);
  const int waves = Bn * (NCHUNK - 1) * 8;                     // 65280 tiles
  dema_pass3<<<waves / 8, 256, 0, stream>>>(x, res, ma, zst, cn);
}